// AutoEncoder_46789373723008
// MI455X (gfx1250) — compile-verified
//
#include <hip/hip_runtime.h>
#include <math.h>

// ---------------------------------------------------------------------------
// Problem constants (from the reference)
// ---------------------------------------------------------------------------
#define BATCH 256
#define GENES 5000
#define G4    (GENES / 4)      // 1250 float4 per gene row
#define FEAT  64
#define NCLS  16
#define NBAT  8

// Output slice offsets (floats), in reference return order
#define OFF_GE  0                              // [256,5000]
#define OFF_Z   (OFF_GE + BATCH*GENES)         // [256,64]
#define OFF_BP  (OFF_Z  + BATCH*FEAT)          // [256,8]
#define OFF_CP  (OFF_BP + BATCH*NBAT)          // [256,16]
#define OFF_SIG (OFF_CP + BATCH*NCLS)          // [64,5000]
#define OFF_INT (OFF_SIG + FEAT*GENES)         // [256,64,5000]

typedef __attribute__((ext_vector_type(2))) float v2f;
typedef __attribute__((ext_vector_type(4))) float v4f;
typedef __attribute__((ext_vector_type(8))) float v8f;

// ---------------------------------------------------------------------------
// Generic f32 GEMM via V_WMMA_F32_16X16X4_F32.
// C[M,N] = act(A[M,K] @ B[K,N] + bias), row-major, K % 4 == 0.
// One wave32 per 16x16 output tile.
//
// Edge handling WITHOUT predication: D(m,n) depends only on A row m and
// B col n, so out-of-range rows/cols are merely CLAMPED for the loads
// (legal addresses, garbage values) — the polluted D rows/cols are never
// stored. Inner loop is branch-free; EXEC is all-1s at every wmma.
//
// Two accumulator chains break the D->C serial dependency between
// consecutive wmmas so loads/wmmas of the two half-steps can overlap.
// ---------------------------------------------------------------------------
__global__ __launch_bounds__(32)
void wmma_gemm_f32(const float* __restrict__ A, const float* __restrict__ B,
                   const float* __restrict__ bias, float* __restrict__ C,
                   int M, int N, int K, int celu)
{
    const int lane = threadIdx.x;
    const int half = lane >> 4;        // 0: lanes 0-15 (K+0,K+1), 1: lanes 16-31 (K+2,K+3)
    const int l16  = lane & 15;
    const int tm   = blockIdx.y;
    const int tn   = blockIdx.x;

    const int rowA = tm * 16 + l16;
    const int colB = tn * 16 + l16;
    const int rA = rowA < M ? rowA : M - 1;   // clamped load row
    const int cB = colB < N ? colB : N - 1;   // clamped load col

    const float* __restrict__ ap = A + (size_t)rA * K + 2 * half;          // contiguous pair
    const float* __restrict__ bp = B + (size_t)(2 * half) * N + cB;        // stride-N pair
    const size_t bstep = (size_t)4 * N;

    v8f acc0 = {}, acc1 = {};
    int k = 0;
    for (; k + 8 <= K; k += 8) {
        v2f a0 = *(const v2f*)ap;              // one global_load_b64
        v2f b0; b0.x = bp[0]; b0.y = bp[N];
        v2f a1 = *(const v2f*)(ap + 4);
        v2f b1; b1.x = bp[bstep]; b1.y = bp[bstep + N];
        acc0 = __builtin_amdgcn_wmma_f32_16x16x4_f32(
                   false, a0, false, b0, (short)0, acc0, false, false);
        acc1 = __builtin_amdgcn_wmma_f32_16x16x4_f32(
                   false, a1, false, b1, (short)0, acc1, false, false);
        ap += 8;
        bp += 2 * bstep;
    }
    for (; k < K; k += 4) {                     // remainder (unused for our dims)
        v2f a = *(const v2f*)ap;
        v2f b; b.x = bp[0]; b.y = bp[N];
        acc0 = __builtin_amdgcn_wmma_f32_16x16x4_f32(
                   false, a, false, b, (short)0, acc0, false, false);
        ap += 4;
        bp += bstep;
    }

    if (colB < N) {
        const float bv = bias ? bias[colB] : 0.0f;
        #pragma unroll
        for (int r = 0; r < 8; ++r) {
            const int row = tm * 16 + r + half * 8;
            if (row < M) {
                float v = acc0[r] + acc1[r] + bv;
                if (celu) v = v > 0.0f ? v : (expf(v) - 1.0f);  // CELU(alpha=1)
                C[(size_t)row * N + colB] = v;
            }
        }
    }
}

// ---------------------------------------------------------------------------
// Discriminator head: 64 -> 64 -> 32 -> OUT, CELU between layers, row softmax.
// One block (64 threads) per batch row; staged in LDS. Negligible work.
// ---------------------------------------------------------------------------
__global__ __launch_bounds__(64)
void disc_mlp_softmax(const float* __restrict__ z,
                      const float* __restrict__ W0, const float* __restrict__ b0,
                      const float* __restrict__ W1, const float* __restrict__ b1,
                      const float* __restrict__ W2, const float* __restrict__ b2,
                      float* __restrict__ out, int OUT)
{
    __shared__ float zs[64], h1[64], h2[32], lg[16];
    const int t = threadIdx.x;
    const int b = blockIdx.x;

    zs[t] = z[b * 64 + t];
    __syncthreads();

    {   // layer 0: 64 -> 64
        float a = b0[t];
        for (int k = 0; k < 64; ++k) a += zs[k] * W0[k * 64 + t];
        h1[t] = a > 0.0f ? a : (expf(a) - 1.0f);
    }
    __syncthreads();

    if (t < 32) {   // layer 1: 64 -> 32
        float a = b1[t];
        for (int k = 0; k < 64; ++k) a += h1[k] * W1[k * 32 + t];
        h2[t] = a > 0.0f ? a : (expf(a) - 1.0f);
    }
    __syncthreads();

    if (t < OUT) {  // layer 2: 32 -> OUT (no activation)
        float a = b2[t];
        for (int k = 0; k < 32; ++k) a += h2[k] * W2[k * OUT + t];
        lg[t] = a;
    }
    __syncthreads();

    if (t < OUT) {  // row softmax (OUT <= 16; redundant per-lane scan is fine)
        float m = -3.0e38f;
        for (int i = 0; i < OUT; ++i) m = fmaxf(m, lg[i]);
        float s = 0.0f;
        for (int i = 0; i < OUT; ++i) s += expf(lg[i] - m);
        out[b * OUT + t] = expf(lg[t] - m) / s;
    }
}

// ---------------------------------------------------------------------------
// Fused:  intermediate[b,f,g] = z[b,f] * prop[f, cls[b], g] * sig[f,g]
//         gene_expression[b,g] = sum_f intermediate + 3*intermediate[b,0,g]
// Store-bandwidth bound: 328 MB of intermediate stores (~14 us at 23.3 TB/s).
// prop's unique working set (20.5 MB) and sig (1.3 MB) live in the 192 MB L2.
// float4 everywhere (G=5000 % 4 == 0, all row offsets 16B-aligned) so the
// dominant stream uses global_store_b128; the F-reduction stays in registers.
// ---------------------------------------------------------------------------
__global__ __launch_bounds__(256)
void fused_intermediate_expr(const float* __restrict__ z,    // [256,64]
                             const float* __restrict__ sig,  // [64,5000]
                             const float* __restrict__ prop, // [64,16,5000]
                             const int*   __restrict__ cls,  // [256]
                             float* __restrict__ inter,      // [256,64,5000]
                             float* __restrict__ ge)         // [256,5000]
{
    const int b  = blockIdx.y;
    const int g4 = blockIdx.x * blockDim.x + threadIdx.x;    // float4 column
    if (g4 >= G4) return;

    const int c = cls[b];
    const float* __restrict__ zrow = z + b * FEAT;
    const v4f* __restrict__ prop4 = (const v4f*)prop;
    const v4f* __restrict__ sig4  = (const v4f*)sig;
    v4f* __restrict__ inter4 = (v4f*)inter;

    v4f acc = {}, first = {};

    for (int f = 0; f < FEAT; ++f) {
        const float zf = zrow[f];
        const v4f p = prop4[((size_t)f * NCLS + c) * G4 + g4];
        const v4f s = sig4[(size_t)f * G4 + g4];
        v4f v;
        v.x = zf * p.x * s.x;
        v.y = zf * p.y * s.y;
        v.z = zf * p.z * s.z;
        v.w = zf * p.w * s.w;
        inter4[((size_t)b * FEAT + f) * G4 + g4] = v;        // global_store_b128
        acc.x += v.x; acc.y += v.y; acc.z += v.z; acc.w += v.w;
        if (f == 0) first = v;
    }

    v4f o;
    o.x = acc.x + 3.0f * first.x;
    o.y = acc.y + 3.0f * first.y;
    o.z = acc.z + 3.0f * first.z;
    o.w = acc.w + 3.0f * first.w;
    ((v4f*)ge)[(size_t)b * G4 + g4] = o;
}

// ---------------------------------------------------------------------------
// Launcher
// ---------------------------------------------------------------------------
static inline dim3 gemm_grid(int M, int N) {
    return dim3((N + 15) / 16, (M + 15) / 16);
}

extern "C" void kernel_launch(void* const* d_in, const int* in_sizes, int n_in,
                              void* d_out, int out_size, void* d_ws, size_t ws_size,
                              hipStream_t stream)
{
    // ---- input map (setup_inputs dict order, tuples flattened) ----
    const float* x    = (const float*)d_in[0];   // [256,5000]
    // d_in[1] expected_sigmatrix: unused by the reference computation
    const int*   cls  = (const int*)  d_in[2];   // [256]
    const float* prop = (const float*)d_in[3];   // [64,16,5000]
    const float* eW[5] = { (const float*)d_in[4],  (const float*)d_in[6],
                           (const float*)d_in[8],  (const float*)d_in[10],
                           (const float*)d_in[12] };
    const float* eB[5] = { (const float*)d_in[5],  (const float*)d_in[7],
                           (const float*)d_in[9],  (const float*)d_in[11],
                           (const float*)d_in[13] };
    const float* dW[5] = { (const float*)d_in[14], (const float*)d_in[15],
                           (const float*)d_in[16], (const float*)d_in[17],
                           (const float*)d_in[18] };
    const float* bdW[3] = { (const float*)d_in[19], (const float*)d_in[21],
                            (const float*)d_in[23] };
    const float* bdB[3] = { (const float*)d_in[20], (const float*)d_in[22],
                            (const float*)d_in[24] };
    const float* cdW[3] = { (const float*)d_in[25], (const float*)d_in[27],
                            (const float*)d_in[29] };
    const float* cdB[3] = { (const float*)d_in[26], (const float*)d_in[28],
                            (const float*)d_in[30] };

    float* out = (float*)d_out;
    float* ge  = out + OFF_GE;
    float* z   = out + OFF_Z;
    float* bp  = out + OFF_BP;
    float* cp  = out + OFF_CP;
    float* sig = out + OFF_SIG;
    float* itm = out + OFF_INT;

    // ---- workspace layout (floats), ~1.2 MB total ----
    float* w  = (float*)d_ws;
    float* t1 = w;              // [64,128]
    float* t2 = t1 + 64 * 128;  // [64,256]
    float* t3 = t2 + 64 * 256;  // [64,512]
    float* h1 = t3 + 64 * 512;  // [256,512]
    float* h2 = h1 + 256 * 512; // [256,256]
    float* h3 = h2 + 256 * 256; // [256,128]
    float* h4 = h3 + 256 * 128; // [256,64]

    // ---- 1) decoder weight chain -> sigmatrix [64,5000] ----
    wmma_gemm_f32<<<gemm_grid(64, 128),  32, 0, stream>>>(dW[0], dW[1], nullptr, t1, 64, 128,  64,  0);
    wmma_gemm_f32<<<gemm_grid(64, 256),  32, 0, stream>>>(t1,    dW[2], nullptr, t2, 64, 256,  128, 0);
    wmma_gemm_f32<<<gemm_grid(64, 512),  32, 0, stream>>>(t2,    dW[3], nullptr, t3, 64, 512,  256, 0);
    wmma_gemm_f32<<<gemm_grid(64, GENES),32, 0, stream>>>(t3,    dW[4], nullptr, sig,64, GENES,512, 0);

    // ---- 2) encoder -> z [256,64] (CELU between layers, none after last) ----
    wmma_gemm_f32<<<gemm_grid(256, 512), 32, 0, stream>>>(x,  eW[0], eB[0], h1, 256, 512, GENES, 1);
    wmma_gemm_f32<<<gemm_grid(256, 256), 32, 0, stream>>>(h1, eW[1], eB[1], h2, 256, 256, 512,  1);
    wmma_gemm_f32<<<gemm_grid(256, 128), 32, 0, stream>>>(h2, eW[2], eB[2], h3, 256, 128, 256,  1);
    wmma_gemm_f32<<<gemm_grid(256, 64),  32, 0, stream>>>(h3, eW[3], eB[3], h4, 256, 64,  128,  1);
    wmma_gemm_f32<<<gemm_grid(256, 64),  32, 0, stream>>>(h4, eW[4], eB[4], z,  256, 64,  64,   0);

    // ---- 3) discriminator heads (softmax rows) ----
    disc_mlp_softmax<<<BATCH, 64, 0, stream>>>(z, bdW[0], bdB[0], bdW[1], bdB[1],
                                               bdW[2], bdB[2], bp, NBAT);
    disc_mlp_softmax<<<BATCH, 64, 0, stream>>>(z, cdW[0], cdB[0], cdW[1], cdB[1],
                                               cdW[2], cdB[2], cp, NCLS);

    // ---- 4) fused intermediate + gene_expression (store-BW bound) ----
    dim3 fgrid((G4 + 255) / 256, BATCH);
    fused_intermediate_expr<<<fgrid, 256, 0, stream>>>(z, sig, prop, cls, itm, ge);
}